// Block_50345606644112
// MI455X (gfx1250) — compile-verified
//
#include <hip/hip_runtime.h>
#include <stdint.h>

// Problem constants (from reference setup_inputs)
#define B_ 8
#define T_ 4096
#define C_ 768
#define CB 256        // channels handled per block (one per thread)
#define NCB (C_ / CB) // 3 channel-blocks per batch
#define TT 16         // timesteps per LDS tile
#define NEG_INF (-1e38f)
#define LOG2E 1.4426950408889634f

#ifdef __has_builtin
#if __has_builtin(__builtin_amdgcn_tensor_load_to_lds)
#define HAVE_TDM 1
#endif
#endif
#ifndef HAVE_TDM
#define HAVE_TDM 0
#endif

// v_exp_f32 is natively exp2 on AMDGCN; this lowers to a single trans op.
__device__ __forceinline__ float ex2(float x) { return __builtin_amdgcn_exp2f(x); }

// ---------------- CDNA5 async global -> LDS (fallback path) ----------------
__device__ __forceinline__ void async_b128(uint32_t lds_off, const float* g) {
  asm volatile("global_load_async_to_lds_b128 %0, %1, off"
               :
               : "v"(lds_off), "v"((uint64_t)(uintptr_t)g)
               : "memory");
}
__device__ __forceinline__ void wait_async0() {
  asm volatile("s_wait_asynccnt 0" ::: "memory");
}

__device__ __forceinline__ void load_tile_async(const float* gk, const float* gv,
                                                int tile, uint32_t ldsK,
                                                uint32_t ldsV, int tid) {
  const int rb  = tid >> 6;         // 0..3
  const int col = (tid & 63) << 2;  // 4 floats per lane
#pragma unroll
  for (int i = 0; i < 4; ++i) {
    const int row = (i << 2) + rb;
    const size_t  goff = (size_t)(tile * TT + row) * C_ + col;
    const uint32_t loff = (uint32_t)((row * CB + col) * 4);
    async_b128(ldsK + loff, gk + goff);
    async_b128(ldsV + loff, gv + goff);
  }
}

// ---------------- CDNA5 Tensor Data Mover (preferred path) ----------------
#if HAVE_TDM
typedef unsigned int v4u __attribute__((ext_vector_type(4)));
typedef int          v4i __attribute__((ext_vector_type(4)));
typedef int          v8i __attribute__((ext_vector_type(8)));

// One 2D tile: TT rows x CB f32 columns, row stride C_ elements.
// D# per CDNA5 ISA (group0: count/lds/global/type,
// group1: data_size, tensor dims, tile dims, dim0 stride). 2D => groups 2/3 NULL.
__device__ __forceinline__ void tdm_load_2d(uint32_t lds_byte_addr, const float* gp) {
  const uint64_t ga = (uint64_t)(uintptr_t)gp;
  v4u g0;
  g0.x = 1u;                                               // count=1 (valid user D#)
  g0.y = lds_byte_addr;                                    // lds_addr
  g0.z = (uint32_t)ga;                                     // global_addr[31:0]
  g0.w = (uint32_t)((ga >> 32) & 0x01FFFFFFu) | (2u << 30);// addr[56:32] | type=2
  v8i g1;
  g1[0] = (int)(2u << 16);             // workgroup_mask=0, data_size=2 (4B)
  g1[1] = (int)((uint32_t)C_ << 16);   // tensor_dim0[15:0] = 768 (bits 63:48)
  g1[2] = (int)((uint32_t)T_ << 16);   // dim0 hi=0 | tensor_dim1[15:0] = 4096
  g1[3] = (int)((uint32_t)CB << 16);   // dim1 hi=0 | tile_dim0 = 256
  g1[4] = (int)TT;                     // tile_dim1 = 16, tile_dim2 = 0
  g1[5] = (int)C_;                     // tensor_dim0_stride[31:0] = 768
  g1[6] = 0;                           // stride hi | tensor_dim1_stride lo
  g1[7] = 0;
  const v4i z4 = {0, 0, 0, 0};
#if __clang_major__ >= 23
  const v8i z8 = {0, 0, 0, 0, 0, 0, 0, 0};
  __builtin_amdgcn_tensor_load_to_lds(g0, g1, z4, z4, z8, 0);
#else
  __builtin_amdgcn_tensor_load_to_lds(g0, g1, z4, z4, 0);
#endif
}

__device__ __forceinline__ void wait_tensor0() {
#if __has_builtin(__builtin_amdgcn_s_wait_tensorcnt)
  __builtin_amdgcn_s_wait_tensorcnt(0);
#else
  asm volatile("s_wait_tensorcnt 0" ::: "memory");
#endif
}
#endif  // HAVE_TDM

// Stage one (TT x CB) tile of k and of v into LDS; TDM if available.
__device__ __forceinline__ void stage_tile(const float* gk, const float* gv,
                                           int tile, uint32_t ldsK, uint32_t ldsV,
                                           int tid) {
#if HAVE_TDM
  if (tid < 32) {  // wave 0 issues both TDM descriptors (EXEC ignored by TDM)
    tdm_load_2d(ldsK, gk + (size_t)tile * TT * C_);
    tdm_load_2d(ldsV, gv + (size_t)tile * TT * C_);
  }
#else
  load_tile_async(gk, gv, tile, ldsK, ldsV, tid);
#endif
}

__device__ __forceinline__ void stage_wait(int tid) {
#if HAVE_TDM
  if (tid < 32) wait_tensor0();  // TENSORcnt is per-wave; only wave 0 issued
#else
  wait_async0();                 // each wave waits for its own async loads
#endif
}

// ---------------- Pass 1: per-chunk transfer summaries ----------------
// State-update half of the WKV step from a zero state, in base-2 log domain
// (scaling all log-quantities by log2(e) preserves the weights exactly and
// drops one v_mul per exp).
template <int S>
__global__ __launch_bounds__(256) void wkv_pass1(
    const float* __restrict__ w, const float* __restrict__ k,
    const float* __restrict__ v, float* __restrict__ sumA,
    float* __restrict__ sumB, float* __restrict__ sumP) {
  __shared__ float lds[2 * 2 * TT * CB];  // 2 buffers x (k,v) x 16x256 = 64KB
  constexpr int L = T_ / S;
  constexpr int NT = L / TT;
  const int tid = threadIdx.x;
  const int s  = blockIdx.x % S;  // power of two -> AND
  const int bc = blockIdx.x / S;
  const int b = bc / NCB, cb = bc % NCB;
  const int c0 = cb * CB, c = c0 + tid;

  const size_t gbase = ((size_t)b * T_ + (size_t)s * L) * C_ + c0;
  const float* gk = k + gbase;
  const float* gv = v + gbase;
  const uint32_t lds0 = (uint32_t)(uintptr_t)&lds[0];

  const float wc = w[c] * LOG2E;
  float aa = 0.f, bb = 0.f, pp = NEG_INF;

  stage_tile(gk, gv, 0, lds0, lds0 + TT * CB * 4, tid);
  stage_wait(tid);
  __syncthreads();

  int buf = 0;
  for (int t = 0; t < NT; ++t) {
    if (t + 1 < NT) {
      const uint32_t nb = lds0 + (uint32_t)(((buf ^ 1) * 2 * TT * CB) * 4);
      stage_tile(gk, gv, t + 1, nb, nb + TT * CB * 4, tid);
    }
    const float* kb = &lds[buf * 2 * TT * CB];
    const float* vb = kb + TT * CB;
#pragma unroll
    for (int tt = 0; tt < TT; ++tt) {
      const float kt = kb[tt * CB + tid] * LOG2E;
      const float vt = vb[tt * CB + tid];
      const float ww2 = wc + pp;
      const float p2  = fmaxf(ww2, kt);
      const float e1b = ex2(ww2 - p2);
      const float e2b = ex2(kt - p2);
      aa = e1b * aa + e2b * vt;
      bb = e1b * bb + e2b;
      pp = p2;
    }
    stage_wait(tid);
    __syncthreads();
    buf ^= 1;
  }
  const size_t idx = (size_t)s * (B_ * C_) + (size_t)b * C_ + c;
  sumA[idx] = aa; sumB[idx] = bb; sumP[idx] = pp;
}

// ---------------- Pass 2: sequential prefix-combine over chunks ----------------
template <int S>
__global__ __launch_bounds__(256) void wkv_pass2(
    const float* __restrict__ w, const float* __restrict__ sumA,
    const float* __restrict__ sumB, const float* __restrict__ sumP,
    float* __restrict__ startA, float* __restrict__ startB,
    float* __restrict__ startP) {
  const int idx = blockIdx.x * blockDim.x + threadIdx.x;  // = b*C + c
  if (idx >= B_ * C_) return;
  const int c = idx % C_;  // constant divisor -> magic multiply
  const float wL = w[c] * (LOG2E * (float)(T_ / S));  // chunk decay, base-2 log
  float aa = 0.f, bb = 0.f, pp = NEG_INF;
#pragma unroll
  for (int s = 0; s < S; ++s) {
    const size_t o = (size_t)s * (B_ * C_) + idx;
    startA[o] = aa; startB[o] = bb; startP[o] = pp;
    const float as = sumA[o], bs = sumB[o], ps = sumP[o];
    const float pdec = pp + wL;  // decayed log2-scale of incoming state
    const float p  = fmaxf(pdec, ps);
    const float e0 = ex2(pdec - p);
    const float e1 = ex2(ps - p);
    aa = e0 * aa + e1 * as;
    bb = e0 * bb + e1 * bs;
    pp = p;
  }
}

// ---------------- Pass 3: replay each chunk from its true state, emit y ----------------
template <int S>
__global__ __launch_bounds__(256) void wkv_pass3(
    const float* __restrict__ w, const float* __restrict__ u,
    const float* __restrict__ k, const float* __restrict__ v,
    const float* __restrict__ startA, const float* __restrict__ startB,
    const float* __restrict__ startP, float* __restrict__ y) {
  __shared__ float lds[2 * 2 * TT * CB];  // 64KB
  constexpr int L = T_ / S;
  constexpr int NT = L / TT;
  const int tid = threadIdx.x;
  const int s  = blockIdx.x % S;
  const int bc = blockIdx.x / S;
  const int b = bc / NCB, cb = bc % NCB;
  const int c0 = cb * CB, c = c0 + tid;

  const size_t gbase = ((size_t)b * T_ + (size_t)s * L) * C_ + c0;
  const float* gk = k + gbase;
  const float* gv = v + gbase;
  float*       gy = y + gbase;
  const uint32_t lds0 = (uint32_t)(uintptr_t)&lds[0];

  const float wc = w[c] * LOG2E, uc = u[c] * LOG2E;
  const size_t sidx = (size_t)s * (B_ * C_) + (size_t)b * C_ + c;
  float aa = startA[sidx], bb = startB[sidx], pp = startP[sidx];

  stage_tile(gk, gv, 0, lds0, lds0 + TT * CB * 4, tid);
  stage_wait(tid);
  __syncthreads();

  int buf = 0;
  for (int t = 0; t < NT; ++t) {
    if (t + 1 < NT) {
      const uint32_t nb = lds0 + (uint32_t)(((buf ^ 1) * 2 * TT * CB) * 4);
      stage_tile(gk, gv, t + 1, nb, nb + TT * CB * 4, tid);
    }
    const float* kb = &lds[buf * 2 * TT * CB];
    const float* vb = kb + TT * CB;
#pragma unroll
    for (int tt = 0; tt < TT; ++tt) {
      const float kt = kb[tt * CB + tid] * LOG2E;
      const float vt = vb[tt * CB + tid];
      // output half of the reference step
      const float ww = uc + kt;
      const float p  = fmaxf(pp, ww);
      const float e1 = ex2(pp - p);
      const float e2 = ex2(ww - p);
      gy[(size_t)(t * TT + tt) * C_ + tid] =
          (e1 * aa + e2 * vt) / (e1 * bb + e2);
      // state half
      const float ww2 = wc + pp;
      const float p2  = fmaxf(ww2, kt);
      const float e1b = ex2(ww2 - p2);
      const float e2b = ex2(kt - p2);
      aa = e1b * aa + e2b * vt;
      bb = e1b * bb + e2b;
      pp = p2;
    }
    stage_wait(tid);
    __syncthreads();
    buf ^= 1;
  }
}

// ---------------- Host-side dispatch ----------------
template <int S>
static void launch_all(const float* w, const float* u, const float* k,
                       const float* v, float* y, float* wsf, hipStream_t stream) {
  const size_t BC = (size_t)B_ * C_;
  float* sumA = wsf;
  float* sumB = sumA + (size_t)S * BC;
  float* sumP = sumB + (size_t)S * BC;
  float* stA  = sumP + (size_t)S * BC;
  float* stB  = stA + (size_t)S * BC;
  float* stP  = stB + (size_t)S * BC;
  dim3 blk(256);
  if (S > 1)  // S==1: single chunk, summaries unused; pass2 just zeroes starts
    wkv_pass1<S><<<dim3(B_ * NCB * S), blk, 0, stream>>>(w, k, v, sumA, sumB, sumP);
  wkv_pass2<S><<<dim3((B_ * C_ + 255) / 256), blk, 0, stream>>>(w, sumA, sumB, sumP,
                                                                stA, stB, stP);
  wkv_pass3<S><<<dim3(B_ * NCB * S), blk, 0, stream>>>(w, u, k, v, stA, stB, stP, y);
}

extern "C" void kernel_launch(void* const* d_in, const int* in_sizes, int n_in,
                              void* d_out, int out_size, void* d_ws, size_t ws_size,
                              hipStream_t stream) {
  // setup_inputs order: B, T, C (scalars), w, u, k, v
  const float* w = (const float*)d_in[3];
  const float* u = (const float*)d_in[4];
  const float* k = (const float*)d_in[5];
  const float* v = (const float*)d_in[6];
  float* y = (float*)d_out;
  float* wsf = (float*)d_ws;

  // Largest chunk count in {32, 8, 1} whose summaries fit the workspace
  // (needs 6*S*B*C floats). Deterministic for a fixed ws_size.
  const size_t BC = (size_t)B_ * C_ * sizeof(float);
  if (6 * 32 * BC <= ws_size)
    launch_all<32>(w, u, k, v, y, wsf, stream);
  else if (6 * 8 * BC <= ws_size)
    launch_all<8>(w, u, k, v, y, wsf, stream);
  else
    launch_all<1>(w, u, k, v, y, wsf, stream);

  (void)in_sizes; (void)n_in; (void)out_size;
}